// Resample2d_3882650436436
// MI455X (gfx1250) — compile-verified
//
#include <hip/hip_runtime.h>
#include <hip/hip_bf16.h>

// FlowNet2 Resample2d (bilinear warp), NCHW fp32.
//   input1: [B=4, C=64, H=384, W=512]
//   flow:   [B=4, 2,     H,     W]   (dx, dy)
// Memory-bound gather: ~410 MB total traffic -> ~17.6 us at 23.3 TB/s.
// One thread per (b,h,w); coordinates computed once, reused for all 64
// channels. Flow staged through the CDNA5 async global->LDS copy path.
// 2D grid (spatial, batch) so every index derivation is shift/mask only.

#define RS_B 4
#define RS_C 64
#define RS_H 384
#define RS_W 512            // power of two -> h/w via shift/mask
#define RS_HW (RS_H * RS_W) // 196608
#define RS_BLOCK 256

extern "C" __global__ __launch_bounds__(RS_BLOCK)
void resample2d_warp_kernel(const float* __restrict__ img,
                            const float* __restrict__ flow,
                            float* __restrict__ out)
{
    __shared__ float sflow[2 * RS_BLOCK];

    const int b  = blockIdx.y;                          // batch
    const int hw = blockIdx.x * RS_BLOCK + threadIdx.x; // spatial index
    const int h  = hw >> 9;                             // hw / 512
    const int w  = hw & (RS_W - 1);                     // hw % 512

    // ---- stage (dx,dy) via CDNA5 async global->LDS copies (ASYNCcnt) ----
    const float* gdx = flow + (size_t)(2 * b + 0) * RS_HW + hw;
    const float* gdy = flow + (size_t)(2 * b + 1) * RS_HW + hw;
    // low 32 bits of a generic pointer to LDS == LDS byte offset
    unsigned lds_x = (unsigned)(size_t)(&sflow[threadIdx.x]);
    unsigned lds_y = (unsigned)(size_t)(&sflow[RS_BLOCK + threadIdx.x]);
    asm volatile("global_load_async_to_lds_b32 %0, %1, off"
                 :: "v"(lds_x), "v"(gdx) : "memory");
    asm volatile("global_load_async_to_lds_b32 %0, %1, off"
                 :: "v"(lds_y), "v"(gdy) : "memory");
    asm volatile("s_wait_asynccnt 0x0" ::: "memory");
    // each lane reads back only its own slot -> no workgroup barrier needed
    const float dx = sflow[threadIdx.x];
    const float dy = sflow[RS_BLOCK + threadIdx.x];

    // ---- bilinear sample coordinate (border-clamped taps, unclamped frac) --
    const float xf  = (float)w + dx;
    const float yf  = (float)h + dy;
    const float x0f = floorf(xf);
    const float y0f = floorf(yf);
    const float fa  = xf - x0f;            // frac weight in x
    const float fb  = yf - y0f;            // frac weight in y
    const int ix0 = (int)x0f;
    const int iy0 = (int)y0f;
    const int x0 = min(max(ix0,     0), RS_W - 1);
    const int x1 = min(max(ix0 + 1, 0), RS_W - 1);
    const int y0 = min(max(iy0,     0), RS_H - 1);
    const int y1 = min(max(iy0 + 1, 0), RS_H - 1);

    const float w00 = (1.0f - fa) * (1.0f - fb);
    const float w01 = fa * (1.0f - fb);
    const float w10 = (1.0f - fa) * fb;
    const float w11 = fa * fb;

    const int i00 = (y0 << 9) + x0;
    const int i01 = (y0 << 9) + x1;
    const int i10 = (y1 << 9) + x0;
    const int i11 = (y1 << 9) + x1;

    const float* __restrict__ src = img + (size_t)b * RS_C * RS_HW;
    float* __restrict__       dst = out + (size_t)b * RS_C * RS_HW + hw;

    // 64 channels share the four tap indices; stores are wave-coalesced.
    // Unroll 8 -> 32 gather loads in flight per thread.
#pragma unroll 8
    for (int c = 0; c < RS_C; ++c) {
        const float* __restrict__ p = src + c * RS_HW;
        const float v = w00 * p[i00] + w01 * p[i01]
                      + w10 * p[i10] + w11 * p[i11];
        dst[(size_t)c * RS_HW] = v;
    }
}

extern "C" void kernel_launch(void* const* d_in, const int* in_sizes, int n_in,
                              void* d_out, int out_size, void* d_ws, size_t ws_size,
                              hipStream_t stream)
{
    (void)in_sizes; (void)n_in; (void)out_size; (void)d_ws; (void)ws_size;
    const float* img  = (const float*)d_in[0];   // input1 [B,C,H,W]
    const float* flow = (const float*)d_in[1];   // input2 [B,2,H,W]
    float* out = (float*)d_out;

    dim3 grid(RS_HW / RS_BLOCK, RS_B);           // (768, 4)
    hipLaunchKernelGGL(resample2d_warp_kernel,
                       grid, dim3(RS_BLOCK), 0, stream,
                       img, flow, out);
}